// TinyCscBottleneck_67894843015850
// MI455X (gfx1250) — compile-verified
//
#include <hip/hip_runtime.h>

// MI455X / gfx1250: wave32, WMMA (V_WMMA_F32_16X16X4_F32), 23.3 TB/s HBM,
// 192 MB L2. Workload is HBM-bound (~390 MiB moved vs ~2 GFLOP):
//   - fp32 throughout (matches reference numerics),
//   - z read exactly twice (pass1: stats + down-proj fused; pass2: epilogue),
//   - z fits in L2 (128 MiB < 192 MB): pass1 z loads are regular-temporal to
//     populate L2; pass2 z loads (last use) and out stores (write-once) are
//     non-temporal so they don't thrash the resident z stream.
//   - GEMMs (C=1024 <-> M=8 per pixel) run on the matrix pipe via fp32 WMMA.

typedef __attribute__((ext_vector_type(2))) float v2f;   // A/B operand: 16x4 / 4x16 f32
typedef __attribute__((ext_vector_type(8))) float v8f;   // C/D operand: 16x16 f32

#define B_   8
#define C_   1024
#define HW_  4096
#define M_   8
#define GAIN 0.3f
#define EPS_ 1e-6f

// d_ws float layout:
//   [0..7]   per-batch sum(z)
//   [8..15]  per-batch sum(z^2)
//   [16..23] Wsum[m] = sum_c W[m][c]
//   [32 ...] yraw[b][m][p]  (B*M*HW = 262144 floats, ~1 MiB)

static __device__ __forceinline__ v8f wmma_f32(v2f a, v2f b, v8f c) {
  return __builtin_amdgcn_wmma_f32_16x16x4_f32(false, a, false, b, (short)0, c,
                                               false, false);
}

// ---------------------------------------------------------------- init ----
__global__ void __launch_bounds__(256) csc_init(const float* __restrict__ Wd,
                                                float* __restrict__ ws) {
  const int tid  = threadIdx.x;
  const int w    = tid >> 5;   // wave id == m row
  const int lane = tid & 31;
  if (tid < 16) ws[tid] = 0.0f;  // zero sum/sumsq accumulators
  if (w < M_) {
    float s = 0.0f;
    for (int i = lane; i < C_; i += 32) s += Wd[w * C_ + i];
    for (int off = 16; off >= 1; off >>= 1) s += __shfl_xor(s, off, 32);
    if (lane == 0) ws[16 + w] = s;
  }
}

// --------------------------------------------------------------- pass 1 ---
// One wave per 16-pixel tile. D(16x16) += A(Wpad 16x4) * B(z 4x16) over 256
// K-steps; same z registers feed sum/sumsq accumulation (single read of z,
// regular-temporal so the stream lands in L2 for pass 2).
__global__ void __launch_bounds__(256) csc_pass1(const float* __restrict__ z,
                                                 const float* __restrict__ Wd,
                                                 float* __restrict__ ws) {
  __shared__ float WTP[C_ * 16];   // W^T padded to 16 m-rows (zeros for m>=8)

  const int tid = threadIdx.x;
  const int b   = blockIdx.y;

  for (int c = tid; c < C_; c += 256) {
#pragma unroll
    for (int mr = 0; mr < 16; ++mr)
      WTP[c * 16 + mr] = (mr < M_) ? Wd[mr * C_ + c] : 0.0f;
  }
  __syncthreads();

  const int wave = tid >> 5;
  const int lane = tid & 31;
  const int hl   = lane >> 4;        // 0: K=0/1, 1: K=2/3 (ISA A/B layout)
  const int l15  = lane & 15;
  const int p0   = (blockIdx.x * 8 + wave) * 16;   // pixel tile base
  const float* zb = z + (size_t)b * C_ * HW_;

  v8f acc = {};
  float s1 = 0.0f, s2 = 0.0f;

  for (int c0 = 0; c0 < C_; c0 += 4) {
    const int ca = c0 + 2 * hl;
    v2f a, bb;
    a.x = WTP[ca * 16 + l15];                 // A[M=l15][K=2*hl]
    a.y = WTP[(ca + 1) * 16 + l15];           // A[M=l15][K=2*hl+1]
    const float z0 = zb[(size_t)ca * HW_ + p0 + l15];        // B[K][N=l15]
    const float z1 = zb[(size_t)(ca + 1) * HW_ + p0 + l15];
    bb.x = z0;
    bb.y = z1;
    acc = wmma_f32(a, bb, acc);
    s1 += z0 + z1;
    s2 = fmaf(z0, z0, s2);
    s2 = fmaf(z1, z1, s2);
  }

  // D layout: acc[g] on lanes 0..15 == D[m=g][pixel l15]; rows 8..15 are pad.
  float* yraw = ws + 32 + (size_t)b * M_ * HW_;
  if (hl == 0) {
#pragma unroll
    for (int g = 0; g < M_; ++g)
      yraw[g * HW_ + p0 + l15] = acc[g];
  }

  for (int off = 16; off >= 1; off >>= 1) {
    s1 += __shfl_xor(s1, off, 32);
    s2 += __shfl_xor(s2, off, 32);
  }
  if (lane == 0) {
    atomicAdd(&ws[b], s1);
    atomicAdd(&ws[8 + b], s2);
  }
}

// --------------------------------------------------------------- pass 2 ---
// r(16c x 16p) = A(W^T 16c x 4m) * B(y 4m x 16p), two WMMAs cover m=0..7.
// Fused epilogue: out = z + GAIN * r. z loads are last-use and out stores are
// write-once -> both non-temporal to preserve z residency in L2.
__global__ void __launch_bounds__(256) csc_pass2(const float* __restrict__ z,
                                                 const float* __restrict__ Wd,
                                                 const float* __restrict__ ws,
                                                 float* __restrict__ out) {
  const int tid  = threadIdx.x;
  const int b    = blockIdx.y;
  const int wave = tid >> 5;
  const int lane = tid & 31;
  const int hl   = lane >> 4;
  const int l15  = lane & 15;
  const int p0   = (blockIdx.x * 8 + wave) * 16;

  const float invN = 1.0f / (float)(C_ * HW_);
  const float mu   = ws[b] * invN;
  const float var  = ws[8 + b] * invN - mu * mu;
  const float rsig = rsqrtf(var + EPS_);

  // B operands: y[m][p] = rsig * (yraw[m][p] - mu * Wsum[m])
  const float* yraw = ws + 32 + (size_t)b * M_ * HW_;
  const int m0 = 2 * hl;                        // lanes 16..31 carry K=2/3
  v2f b1, b2;
  b1.x = rsig * (yraw[(m0 + 0) * HW_ + p0 + l15] - mu * ws[16 + m0 + 0]);
  b1.y = rsig * (yraw[(m0 + 1) * HW_ + p0 + l15] - mu * ws[16 + m0 + 1]);
  b2.x = rsig * (yraw[(m0 + 4) * HW_ + p0 + l15] - mu * ws[16 + m0 + 4]);
  b2.y = rsig * (yraw[(m0 + 5) * HW_ + p0 + l15] - mu * ws[16 + m0 + 5]);

  const float* zb = z + (size_t)b * C_ * HW_;
  float* ob       = out + (size_t)b * C_ * HW_;

  for (int cb = 0; cb < C_; cb += 16) {
    v2f a1, a2;                                  // A[c_row=l15][K=m]
    a1.x = Wd[(m0 + 0) * C_ + cb + l15];
    a1.y = Wd[(m0 + 1) * C_ + cb + l15];
    a2.x = Wd[(m0 + 4) * C_ + cb + l15];
    a2.y = Wd[(m0 + 5) * C_ + cb + l15];

    v8f d = {};
    d = wmma_f32(a1, b1, d);   // m = 0..3
    d = wmma_f32(a2, b2, d);   // m = 4..7

    // d[g] == D[M = g + 8*hl][N = l15] -> channel cb+g+8*hl, pixel p0+l15
    const size_t base = (size_t)(cb + 8 * hl) * HW_ + p0 + l15;
#pragma unroll
    for (int g = 0; g < 8; ++g) {
      const float zv = __builtin_nontemporal_load(zb + base + (size_t)g * HW_);
      __builtin_nontemporal_store(fmaf(GAIN, d[g], zv),
                                  ob + base + (size_t)g * HW_);
    }
  }
}

// ------------------------------------------------------------- launcher ---
extern "C" void kernel_launch(void* const* d_in, const int* in_sizes, int n_in,
                              void* d_out, int out_size, void* d_ws, size_t ws_size,
                              hipStream_t stream) {
  const float* z  = (const float*)d_in[0];   // (8,1024,64,64) fp32
  const float* Wd = (const float*)d_in[1];   // (8,1024) fp32
  float* out = (float*)d_out;
  float* ws  = (float*)d_ws;                 // needs ~1.05 MB

  csc_init<<<dim3(1), dim3(256), 0, stream>>>(Wd, ws);
  dim3 grid(32, B_);                         // 32 WGs x 8 waves = 256 tiles/batch
  csc_pass1<<<grid, dim3(256), 0, stream>>>(z, Wd, ws);
  csc_pass2<<<grid, dim3(256), 0, stream>>>(z, Wd, ws, out);
}